// GATBased_90202903150881
// MI455X (gfx1250) — compile-verified
//
#include <hip/hip_runtime.h>
#include <math.h>

#define NN 1024
#define EE 65536

typedef __attribute__((ext_vector_type(2))) float v2f;
typedef __attribute__((ext_vector_type(8))) float v8f;

// ---------------- workspace layout (float offsets) ----------------
static const size_t OFF_MAX1 = 0;        // N*8 u32  (enc(-inf) filled)
static const size_t OFF_DEN1 = 8192;     // N*8 f32  (zero)
static const size_t OFF_TN1  = 16384;    // N*8 f32  (zero)
static const size_t OFF_MAX2 = 24576;    // N*8 u32
static const size_t OFF_DEN2 = 32768;    // N*8 f32  (zero)
static const size_t OFF_AS2  = 40960;    // N*8
static const size_t OFF_AD2  = 49152;    // N*8
static const size_t OFF_SC   = 57344;    // 32: s1[8] d1[8] se1[8] se2[8]
static const size_t OFF_MM   = 57376;    // 2 u32: min,max bits of dist
static const size_t OFF_MASK = 57408;    // 1024 int (zero)
static const size_t OFF_KVEC = 58432;    // 1024 (zero, atomic-accumulated)
static const size_t OFF_X1   = 65536;    // N*256
static const size_t OFF_H2   = 327680;   // N*256
static const size_t OFF_O2   = 589824;   // N*256 (zero)
static const size_t OFF_X2   = 851968;   // N*256
static const size_t OFF_EA   = 1114112;  // N*32
static const size_t OFF_EB   = 1146880;  // N*32
static const size_t OFF_S    = 1179648;  // 32*1024
static const size_t OFF_P2   = 1212416;  // 32*1024
static const size_t OFF_OUT1 = 1245184;  // N*N

// ordered-uint encoding of float (monotonic for all finite floats)
__device__ __forceinline__ unsigned encf(float f) {
  unsigned u = __float_as_uint(f);
  return (u & 0x80000000u) ? ~u : (u | 0x80000000u);
}
__device__ __forceinline__ float decf(unsigned u) {
  return (u & 0x80000000u) ? __uint_as_float(u & 0x7FFFFFFFu) : __uint_as_float(~u);
}

// ---------------- WMMA f32 16x16 tile GEMM helpers ----------------
// A-frag (16x4 f32): lane l, reg r -> A[M = l&15, K = r + ((l>=16)?2:0)]
// B-frag (4x16 f32): lane l, reg r -> B[K = r + ((l>=16)?2:0), N = l&15]
// C/D (16x16 f32):   lane l, reg r -> C[M = r + ((l>=16)?8:0), N = l&15]
__device__ __forceinline__ void gemm_acc(const float* __restrict__ A, int lda,
                                         const float* __restrict__ B, int ldb,
                                         int kdim, float scaleA,
                                         int m0, int n0, v8f& acc) {
  int lane = threadIdx.x & 31;
  int r    = lane & 15;
  int khi  = (lane >> 4) << 1;  // 0 or 2
  const float* Arow = A + (size_t)(m0 + r) * lda + khi;
  const float* Bp   = B + (size_t)khi * ldb + n0 + r;
  for (int k = 0; k < kdim; k += 4) {
    v2f a, b;
    a[0] = Arow[k] * scaleA;
    a[1] = Arow[k + 1] * scaleA;
    b[0] = Bp[(size_t)k * ldb];
    b[1] = Bp[(size_t)k * ldb + ldb];
    acc = __builtin_amdgcn_wmma_f32_16x16x4_f32(false, a, false, b,
                                                (short)0, acc, false, false);
  }
}

__device__ __forceinline__ void store_tile(float* __restrict__ C, int ldc,
                                           int m0, int n0, const v8f& acc) {
  int lane = threadIdx.x & 31;
  int col  = lane & 15;
  int mo   = (lane >> 4) << 3;
  for (int r = 0; r < 8; ++r)
    C[(size_t)(m0 + mo + r) * ldc + n0 + col] = acc[r];
}

// ---------- LDS-staged 64x64-per-block GEMM (K=1024, chunks of 32) ----------
// LDS B panel: [32 k][72 floats] padded stride -> the khi=2 half-wave is skewed
// by 144 floats == 16 banks, so lanes 0-15 and 16-31 hit disjoint bank sets.
#define KCH   32
#define BSTR  72
#define BUFSZ (KCH * BSTR)

// stage one 32(K) x 64(N) chunk of B into LDS buffer via async DMA
__device__ __forceinline__ void stage_chunk(const float* __restrict__ B, int ldb,
                                            int kc, int n0, float* sB, int buf) {
  int t  = threadIdx.x;            // 0..127
  int r  = t >> 2;                 // K row 0..31
  int cg = (t & 3) << 4;           // col group *16 floats
  unsigned lds  = (unsigned)(uintptr_t)(sB + buf * BUFSZ + r * BSTR + cg);
  unsigned goff = (unsigned)(((kc + r) * ldb + n0 + cg) * 4);
#pragma unroll
  for (int i = 0; i < 4; ++i) {    // 4 x b128 = 16 floats per thread
    asm volatile("global_load_async_to_lds_b128 %0, %1, %2 offset:0"
                 :
                 : "v"(lds + (unsigned)(i * 16)), "v"(goff + (unsigned)(i * 16)),
                   "s"(B)
                 : "memory");
  }
}

// one wave: 16(M) x 64(N), A from global (reused x4), B from LDS
__device__ __forceinline__ void compute_chunk(const float* __restrict__ A, int lda,
                                              float scaleA, int m0, int kc,
                                              const float* sbuf, v8f acc[4]) {
  int lane = threadIdx.x & 31;
  int r    = lane & 15;
  int khi  = (lane >> 4) << 1;
  const float* Arow = A + (size_t)(m0 + r) * lda + kc + khi;
#pragma unroll
  for (int k = 0; k < KCH; k += 4) {
    v2f a;
    a[0] = Arow[k] * scaleA;
    a[1] = Arow[k + 1] * scaleA;
    const float* b0 = sbuf + (k + khi) * BSTR + r;
    const float* b1 = b0 + BSTR;
#pragma unroll
    for (int nt = 0; nt < 4; ++nt) {
      v2f b;
      b[0] = b0[nt * 16];
      b[1] = b1[nt * 16];
      acc[nt] = __builtin_amdgcn_wmma_f32_16x16x4_f32(false, a, false, b,
                                                      (short)0, acc[nt], false, false);
    }
  }
}

// full K=1024 phase with double-buffered async pipeline
__device__ __forceinline__ void gemm_phase_lds(const float* __restrict__ A, int lda,
                                               const float* __restrict__ B, int ldb,
                                               float scaleA, int m0, int n0,
                                               float* sB, v8f acc[4]) {
  stage_chunk(B, ldb, 0, n0, sB, 0);
  for (int c = 0; c < NN / KCH; ++c) {
    asm volatile("s_wait_asynccnt 0x0" ::: "memory");
    __syncthreads();                               // whole panel chunk arrived
    if (c + 1 < NN / KCH) stage_chunk(B, ldb, (c + 1) * KCH, n0, sB, (c + 1) & 1);
    compute_chunk(A, lda, scaleA, m0, c * KCH, sB + (c & 1) * BUFSZ, acc);
    __syncthreads();                               // done reading before reuse
  }
}

// ---------------- small kernels ----------------
__global__ void k_fillmax(unsigned* m1, unsigned* m2) {
  int i = blockIdx.x * blockDim.x + threadIdx.x;
  if (i < 8192) { m1[i] = 0x007FFFFFu; m2[i] = 0x007FFFFFu; }  // enc(-inf)
}

__global__ void k_minmax(const float* __restrict__ dist, unsigned* mm) {
  int tid = blockIdx.x * blockDim.x + threadIdx.x;
  int stride = gridDim.x * blockDim.x;
  unsigned lmin = 0xFFFFFFFFu, lmax = 0u;
  for (int i = tid; i < NN * NN; i += stride) {
    unsigned u = __float_as_uint(dist[i]);  // dist >= 0 -> bit order == float order
    lmin = lmin < u ? lmin : u;
    lmax = lmax > u ? lmax : u;
  }
  atomicMin(&mm[0], lmin);
  atomicMax(&mm[1], lmax);
}

// sc[0..7]=s1  sc[8..15]=d1  sc[16..23]=se1  sc[24..31]=se2
__global__ void k_small(const float* W1, const float* We1, const float* We2,
                        const float* as1, const float* ad1, const float* ae1,
                        const float* ae2, float* sc) {
  int t = threadIdx.x;
  if (t >= 32) return;
  int g = t >> 3, h = t & 7;
  const float* wv = (g < 2) ? W1 : (g == 2 ? We1 : We2);
  const float* av = (g == 0) ? as1 : (g == 1 ? ad1 : (g == 2 ? ae1 : ae2));
  float s = 0.f;
  for (int c = 0; c < 32; ++c) s += wv[h * 32 + c] * av[h * 32 + c];
  sc[t] = s;
}

__global__ void k_mask(const int* __restrict__ stops, int* __restrict__ mask) {
  int i = blockIdx.x * blockDim.x + threadIdx.x;
  if (i < 512) mask[stops[i]] = 1;
}

// ---------------- GAT layer 1 (h = demand ⊗ W1 => scalar per (node,head)) ----
__global__ void k_l1a(const int* __restrict__ ei, const float* __restrict__ markov,
                      const float* __restrict__ demand, const float* __restrict__ sc,
                      unsigned* __restrict__ max1) {
  int idx = blockIdx.x * blockDim.x + threadIdx.x;
  if (idx >= EE * 8) return;
  int e = idx >> 3, h = idx & 7;
  int s = ei[e], d = ei[EE + e];
  float me = markov[s * NN + d];
  float a = demand[s] * sc[h] + demand[d] * sc[8 + h] + me * sc[16 + h];
  a = a > 0.f ? a : 0.2f * a;  // leaky_relu 0.2
  atomicMax(&max1[d * 8 + h], encf(a));
}

__global__ void k_l1b(const int* __restrict__ ei, const float* __restrict__ markov,
                      const float* __restrict__ demand, const float* __restrict__ sc,
                      const unsigned* __restrict__ max1,
                      float* __restrict__ den1, float* __restrict__ tn1) {
  int idx = blockIdx.x * blockDim.x + threadIdx.x;
  if (idx >= EE * 8) return;
  int e = idx >> 3, h = idx & 7;
  int s = ei[e], d = ei[EE + e];
  float me = markov[s * NN + d];
  float a = demand[s] * sc[h] + demand[d] * sc[8 + h] + me * sc[16 + h];
  a = a > 0.f ? a : 0.2f * a;
  float ex = expf(a - decf(max1[d * 8 + h]));
  atomicAdd(&den1[d * 8 + h], ex);
  atomicAdd(&tn1[d * 8 + h], ex * demand[s]);
}

__global__ void k_x1(const float* __restrict__ W1, const float* __restrict__ b1,
                     const float* __restrict__ den1, const float* __restrict__ tn1,
                     float* __restrict__ x1) {
  int idx = blockIdx.x * blockDim.x + threadIdx.x;  // N*256
  int n = idx >> 8, j = idx & 255, h = j >> 5;
  float coef = tn1[n * 8 + h] / (den1[n * 8 + h] + 1e-16f);
  float v = W1[j] * coef + b1[j];
  x1[idx] = v > 0.f ? v : 0.f;
}

// ---------------- h2 = x1 @ W2 (WMMA, small, L2-resident) ----------------
__global__ void k_gemm_h2(const float* __restrict__ x1, const float* __restrict__ W2,
                          float* __restrict__ h2) {
  int wave = (blockIdx.x * blockDim.x + threadIdx.x) >> 5;  // 1024 tiles: 64(M) x 16(N)
  int tm = wave >> 4, tn = wave & 15;
  v8f acc = {};
  gemm_acc(x1, 256, W2, 256, 256, 1.f, tm * 16, tn * 16, acc);
  store_tile(h2, 256, tm * 16, tn * 16, acc);
}

__global__ void k_asad2(const float* __restrict__ h2, const float* __restrict__ as,
                        const float* __restrict__ ad, float* __restrict__ as2,
                        float* __restrict__ ad2) {
  int idx = blockIdx.x * blockDim.x + threadIdx.x;  // 2*N*8
  int which = idx >= NN * 8;
  int t = idx - which * NN * 8;
  int n = t >> 3, h = t & 7;
  const float* av = which ? ad : as;
  float s = 0.f;
  for (int c = 0; c < 32; ++c) s += h2[n * 256 + h * 32 + c] * av[h * 32 + c];
  (which ? ad2 : as2)[n * 8 + h] = s;
}

// ---------------- GAT layer 2 ----------------
__global__ void k_l2a(const int* __restrict__ ei, const float* __restrict__ markov,
                      const float* __restrict__ as2, const float* __restrict__ ad2,
                      const float* __restrict__ sc, unsigned* __restrict__ max2) {
  int idx = blockIdx.x * blockDim.x + threadIdx.x;
  if (idx >= EE * 8) return;
  int e = idx >> 3, h = idx & 7;
  int s = ei[e], d = ei[EE + e];
  float a = as2[s * 8 + h] + ad2[d * 8 + h] + markov[s * NN + d] * sc[24 + h];
  a = a > 0.f ? a : 0.2f * a;
  atomicMax(&max2[d * 8 + h], encf(a));
}

// one wave per edge: lanes<8 compute exp per head, all lanes scatter C=32 channels
__global__ void k_l2b(const int* __restrict__ ei, const float* __restrict__ markov,
                      const float* __restrict__ as2, const float* __restrict__ ad2,
                      const float* __restrict__ sc, const unsigned* __restrict__ max2,
                      const float* __restrict__ h2,
                      float* __restrict__ den2, float* __restrict__ o2) {
  int wid = (blockIdx.x * blockDim.x + threadIdx.x) >> 5;
  int lane = threadIdx.x & 31;
  if (wid >= EE) return;
  int s = ei[wid], d = ei[EE + wid];
  float ex = 0.f;
  if (lane < 8) {
    float a = as2[s * 8 + lane] + ad2[d * 8 + lane] + markov[s * NN + d] * sc[24 + lane];
    a = a > 0.f ? a : 0.2f * a;
    ex = expf(a - decf(max2[d * 8 + lane]));
    atomicAdd(&den2[d * 8 + lane], ex);
  }
  const float* hrow = h2 + (size_t)s * 256;
  float* orow = o2 + (size_t)d * 256;
  for (int h = 0; h < 8; ++h) {
    float exh = __shfl(ex, h, 32);
    atomicAdd(&orow[h * 32 + lane], exh * hrow[h * 32 + lane]);
  }
}

__global__ void k_x2(const float* __restrict__ o2, const float* __restrict__ den2,
                     const float* __restrict__ b2, float* __restrict__ x2) {
  int idx = blockIdx.x * blockDim.x + threadIdx.x;  // N*256
  int n = idx >> 8, j = idx & 255;
  float v = o2[idx] / (den2[n * 8 + (j >> 5)] + 1e-16f) + b2[j];
  x2[idx] = v > 0.f ? v : 0.f;
}

// ---------------- edge summarizer: ea = x2@Wes_top, eb = x2@Wes_bot ----------
__global__ void k_eab(const float* __restrict__ x2, const float* __restrict__ Wes,
                      float* __restrict__ ea, float* __restrict__ eb) {
  int idx = blockIdx.x * blockDim.x + threadIdx.x;  // N*64
  int n = idx >> 6, t = idx & 63;
  int k = t & 31, which = t >> 5;
  const float* w = Wes + (size_t)which * 256 * 32 + k;
  const float* xr = x2 + (size_t)n * 256;
  float s = 0.f;
  for (int j = 0; j < 256; ++j) s += xr[j] * w[(size_t)j * 32];
  (which ? eb : ea)[n * 32 + k] = s;
}

// S[k,m] = sum_j Wc1[j*32+k, m];  P2[k,m] = sum_j (eb[j,k]+bes[k]) * Wc1[j*32+k, m]
__global__ void k_S(const float* __restrict__ Wc1, const float* __restrict__ eb,
                    const float* __restrict__ bes, float* __restrict__ S,
                    float* __restrict__ P2) {
  int idx = blockIdx.x * blockDim.x + threadIdx.x;  // 32*1024
  int m = idx & 1023, k = idx >> 10;
  const float* base = Wc1 + (size_t)k * 1024 + m;
  float bk = bes[k];
  float s = 0.f, p = 0.f;
  for (int j = 0; j < 1024; ++j) {
    float w = base[(size_t)j * 32768];
    s += w;
    p += (eb[j * 32 + k] + bk) * w;
  }
  S[k * 1024 + m] = s;
  P2[k * 1024 + m] = p;
}

// Kvec[m]: all row-constant contributions of combined@Wc1, plus bc1
__global__ void k_Kvec(const float* __restrict__ Wc1, const float* __restrict__ P2,
                       const int* __restrict__ mask, const unsigned* __restrict__ mm,
                       const float* __restrict__ ew, const float* __restrict__ ec,
                       const float* __restrict__ ev, const int* __restrict__ weekday,
                       const int* __restrict__ capacity, const int* __restrict__ vehicles,
                       const float* __restrict__ bc1, float* __restrict__ Kvec) {
  int idx = blockIdx.x * blockDim.x + threadIdx.x;  // 8*1024
  int m = idx & 1023, p = idx >> 10;
  float dmin = __uint_as_float(mm[0]), dmax = __uint_as_float(mm[1]);
  float inv = 1.f / (dmax - dmin);
  const float* Wd  = Wc1 + (size_t)32768 * 1024 + m;
  const float* Wsf = Wc1 + (size_t)34825 * 1024 + m;
  int j0 = p * 128;
  float cs = 0.f, sfv = 0.f;
  for (int j = j0; j < j0 + 128; ++j) {
    cs += Wd[(size_t)j * 1024];
    if (mask[j]) sfv += Wsf[(size_t)j * 1024];
  }
  float add = cs * (-dmin * inv) + sfv;
  if (p == 0) {
    float c2 = 0.f;
    for (int k = 0; k < 32; ++k) c2 += P2[k * 1024 + m];
    const float* Ww = Wc1 + (size_t)34816 * 1024 + m;
    int wd = weekday[0], cp = capacity[0], vh = vehicles[0];
    float fc = 0.f;
    for (int t = 0; t < 3; ++t) {
      fc += ew[wd * 3 + t] * Ww[(size_t)t * 1024];
      fc += ec[cp * 3 + t] * Ww[(size_t)(3 + t) * 1024];
      fc += ev[vh * 3 + t] * Ww[(size_t)(6 + t) * 1024];
    }
    add += c2 + fc + bc1[m];
  }
  atomicAdd(&Kvec[m], add);
}

// out1 = relu(inv*dist@Wc1_d + markov@Wc1_m + ea@S + Kvec)   [async-LDS staged]
__global__ void __launch_bounds__(128)
k_gemm_comb(const float* __restrict__ dist, const float* __restrict__ markov,
            const float* __restrict__ ea, const float* __restrict__ S,
            const float* __restrict__ Wc1, const float* __restrict__ Kvec,
            const unsigned* __restrict__ mm, float* __restrict__ out1) {
  __shared__ float sB[2 * BUFSZ];
  int blk = blockIdx.x;                 // 256 blocks; 64x64 C tile each
  int tmb = blk >> 4, tnb = blk & 15;
  int w = threadIdx.x >> 5;
  int m0 = tmb * 64 + w * 16, n0 = tnb * 64;
  float inv = 1.f / (__uint_as_float(mm[1]) - __uint_as_float(mm[0]));
  v8f acc[4] = {};
  gemm_phase_lds(dist,   NN, Wc1 + (size_t)32768 * NN, NN, inv, m0, n0, sB, acc);
  gemm_phase_lds(markov, NN, Wc1 + (size_t)33792 * NN, NN, 1.f, m0, n0, sB, acc);
#pragma unroll
  for (int nt = 0; nt < 4; ++nt)
    gemm_acc(ea, 32, S, NN, 32, 1.f, m0, n0 + nt * 16, acc[nt]);
  int lane = threadIdx.x & 31;
  int col = lane & 15, mo = (lane >> 4) << 3;
#pragma unroll
  for (int nt = 0; nt < 4; ++nt) {
    float kv = Kvec[n0 + nt * 16 + col];
    for (int r = 0; r < 8; ++r) {
      float v = acc[nt][r] + kv;
      out1[(size_t)(m0 + mo + r) * NN + n0 + nt * 16 + col] = v > 0.f ? v : 0.f;
    }
  }
}

// out = out1 @ Wc2 + bc2   [async-LDS staged]
__global__ void __launch_bounds__(128)
k_gemm_out(const float* __restrict__ out1, const float* __restrict__ Wc2,
           const float* __restrict__ bc2, float* __restrict__ out) {
  __shared__ float sB[2 * BUFSZ];
  int blk = blockIdx.x;                 // 256 blocks; 64x64 C tile each
  int tmb = blk >> 4, tnb = blk & 15;
  int w = threadIdx.x >> 5;
  int m0 = tmb * 64 + w * 16, n0 = tnb * 64;
  v8f acc[4] = {};
  gemm_phase_lds(out1, NN, Wc2, NN, 1.f, m0, n0, sB, acc);
  int lane = threadIdx.x & 31;
  int col = lane & 15, mo = (lane >> 4) << 3;
#pragma unroll
  for (int nt = 0; nt < 4; ++nt) {
    float b = bc2[n0 + nt * 16 + col];
    for (int r = 0; r < 8; ++r)
      out[(size_t)(m0 + mo + r) * NN + n0 + nt * 16 + col] = acc[nt][r] + b;
  }
}

// ---------------- launcher ----------------
extern "C" void kernel_launch(void* const* d_in, const int* in_sizes, int n_in,
                              void* d_out, int out_size, void* d_ws, size_t ws_size,
                              hipStream_t stream) {
  const float* dist    = (const float*)d_in[0];
  const int*   stops   = (const int*)d_in[1];
  const int*   weekday = (const int*)d_in[2];
  const int*   vehicles= (const int*)d_in[3];
  const float* markov  = (const float*)d_in[4];
  const float* demand  = (const float*)d_in[5];
  const int*   capacity= (const int*)d_in[6];
  const int*   ei      = (const int*)d_in[7];
  const float* W1      = (const float*)d_in[8];
  const float* We1     = (const float*)d_in[9];
  const float* as1     = (const float*)d_in[10];
  const float* ad1     = (const float*)d_in[11];
  const float* ae1     = (const float*)d_in[12];
  const float* b1      = (const float*)d_in[13];
  const float* W2      = (const float*)d_in[14];
  const float* We2     = (const float*)d_in[15];
  const float* as_2    = (const float*)d_in[16];
  const float* ad_2    = (const float*)d_in[17];
  const float* ae2     = (const float*)d_in[18];
  const float* b2      = (const float*)d_in[19];
  const float* Wes     = (const float*)d_in[20];
  const float* bes     = (const float*)d_in[21];
  const float* ew      = (const float*)d_in[22];
  const float* ec      = (const float*)d_in[23];
  const float* ev      = (const float*)d_in[24];
  const float* Wc1     = (const float*)d_in[25];
  const float* bc1     = (const float*)d_in[26];
  const float* Wc2     = (const float*)d_in[27];
  const float* bc2     = (const float*)d_in[28];

  float* ws = (float*)d_ws;
  unsigned* max1 = (unsigned*)(ws + OFF_MAX1);
  unsigned* max2 = (unsigned*)(ws + OFF_MAX2);
  unsigned* mm   = (unsigned*)(ws + OFF_MM);
  int* mask      = (int*)(ws + OFF_MASK);

  // zero / fill scratch (graph-capture-safe stream ops only)
  hipMemsetAsync(ws + OFF_DEN1, 0, 2 * 8192 * sizeof(float), stream);  // den1+tn1
  hipMemsetAsync(ws + OFF_DEN2, 0, 8192 * sizeof(float), stream);
  hipMemsetAsync(ws + OFF_O2,   0, 262144 * sizeof(float), stream);
  hipMemsetAsync(ws + OFF_KVEC, 0, 1024 * sizeof(float), stream);
  hipMemsetAsync(ws + OFF_MASK, 0, 1024 * sizeof(int), stream);
  hipMemsetAsync(mm,     0xFF, sizeof(unsigned), stream);  // min = 0xFFFFFFFF
  hipMemsetAsync(mm + 1, 0x00, sizeof(unsigned), stream);  // max = 0

  k_fillmax<<<32, 256, 0, stream>>>(max1, max2);
  k_minmax<<<16, 256, 0, stream>>>(dist, mm);
  k_small<<<1, 32, 0, stream>>>(W1, We1, We2, as1, ad1, ae1, ae2, ws + OFF_SC);
  k_mask<<<2, 256, 0, stream>>>(stops, mask);

  // GAT layer 1
  k_l1a<<<2048, 256, 0, stream>>>(ei, markov, demand, ws + OFF_SC, max1);
  k_l1b<<<2048, 256, 0, stream>>>(ei, markov, demand, ws + OFF_SC, max1,
                                  ws + OFF_DEN1, ws + OFF_TN1);
  k_x1<<<1024, 256, 0, stream>>>(W1, b1, ws + OFF_DEN1, ws + OFF_TN1, ws + OFF_X1);

  // GAT layer 2
  k_gemm_h2<<<256, 128, 0, stream>>>(ws + OFF_X1, W2, ws + OFF_H2);
  k_asad2<<<64, 256, 0, stream>>>(ws + OFF_H2, as_2, ad_2, ws + OFF_AS2, ws + OFF_AD2);
  k_l2a<<<2048, 256, 0, stream>>>(ei, markov, ws + OFF_AS2, ws + OFF_AD2,
                                  ws + OFF_SC, max2);
  k_l2b<<<8192, 256, 0, stream>>>(ei, markov, ws + OFF_AS2, ws + OFF_AD2,
                                  ws + OFF_SC, max2, ws + OFF_H2,
                                  ws + OFF_DEN2, ws + OFF_O2);
  k_x2<<<1024, 256, 0, stream>>>(ws + OFF_O2, ws + OFF_DEN2, b2, ws + OFF_X2);

  // summarizer + combiner constants
  k_eab<<<256, 256, 0, stream>>>(ws + OFF_X2, Wes, ws + OFF_EA, ws + OFF_EB);
  k_S<<<128, 256, 0, stream>>>(Wc1, ws + OFF_EB, bes, ws + OFF_S, ws + OFF_P2);
  k_Kvec<<<32, 256, 0, stream>>>(Wc1, ws + OFF_P2, mask, mm, ew, ec, ev,
                                 weekday, capacity, vehicles, bc1, ws + OFF_KVEC);

  // big fused GEMMs (async-LDS staged WMMA)
  k_gemm_comb<<<256, 128, 0, stream>>>(dist, markov, ws + OFF_EA, ws + OFF_S,
                                       Wc1, ws + OFF_KVEC, mm, ws + OFF_OUT1);
  k_gemm_out<<<256, 128, 0, stream>>>(ws + OFF_OUT1, Wc2, bc2, (float*)d_out);
}